// STULayer2D_48550310314385
// MI455X (gfx1250) — compile-verified
//
#include <hip/hip_runtime.h>
#include <math.h>

// ---------------------------------------------------------------------------
// STU layer on MI455X (gfx1250): entire pipeline as f32 WMMA GEMMs.
//   LN -> rowRFFT(GEMM) -> colDFT(GEMM) -> spectral mix (GEMM, 16 filters)
//      -> inv colDFT(GEMM) -> inv rowFFT-to-real(GEMM) -> fused MLP (GEMM+GELU+GEMM)
// Matrix math: v_wmma_f32_16x16x4_f32 (wave32, 16x16 tiles).
// Tile staging: GLOBAL_LOAD_ASYNC_TO_LDS_B128 (ASYNCcnt) when the toolchain
// declares the builtin; synchronous copy fallback otherwise.
// ---------------------------------------------------------------------------

typedef __attribute__((ext_vector_type(2))) float v2f;
typedef __attribute__((ext_vector_type(8))) float v8f;
typedef int v4i_ __attribute__((vector_size(16)));  // matches builtin's int4 pointee

#define NB   4
#define ND   128
#define NGX  128
#define NGY  128
#define NL   16
#define NYR  65
#define FP   80            // padded half-spectrum width (65 -> 80, pad = zeros)
#define SPW  (NGX * FP)    // 10240 spectral columns per (b, channel)
#define NPIX (NGX * NGY)   // 16384
#define LDA132 132         // LDS A-tile stride (== 4 mod 8 -> conflict-free b64 frags)
#define LDA84  84
#define LDG144 144         // LDS B-table stride (2*144 == 32 mod 64 -> no bank clash)

// table offsets inside ws (floats)
#define T_FYRE 0
#define T_FYIM 10240
#define T_WCF  20480
#define T_WSF  36864
#define T_WCI  53248
#define T_WSI  69632
#define T_GR   86016
#define T_GI   96256
#define T_TOT  106496

// ---------------------------------------------------------------------------
// Async global->LDS staging (CDNA5 ASYNCcnt path), with safe fallback
// ---------------------------------------------------------------------------
#if defined(__has_builtin)
#if __has_builtin(__builtin_amdgcn_global_load_async_to_lds_b128)
#define USE_ASYNC_LDS 1
#endif
#endif
#ifndef USE_ASYNC_LDS
#define USE_ASYNC_LDS 0
#endif

__device__ __forceinline__ void async_join() {
#if USE_ASYNC_LDS
#if __has_builtin(__builtin_amdgcn_s_wait_asynccnt)
  __builtin_amdgcn_s_wait_asynccnt(0);
#else
  asm volatile("s_wait_asynccnt 0" ::: "memory");
#endif
#endif
}

// Copy rows x cols f32 tile (16B-aligned rows) global -> LDS with 256 threads.
__device__ __forceinline__ void stage_tile(float* lds, int ldl, const float* g,
                                           long ldg, int rows, int cols, int tid) {
#if USE_ASYNC_LDS
  int cpr = cols >> 2;  // 16-byte chunks per row
  int total = rows * cpr;
  for (int i = tid; i < total; i += 256) {
    int r = i / cpr, c4 = (i - r * cpr) << 2;
    __builtin_amdgcn_global_load_async_to_lds_b128(
        (v4i_ __attribute__((address_space(1)))*)(g + (long)r * ldg + c4),
        (v4i_ __attribute__((address_space(3)))*)(lds + r * ldl + c4), 0, 0);
  }
#else
  for (int i = tid; i < rows * cols; i += 256) {
    int r = i / cols, c = i - r * cols;
    lds[r * ldl + c] = g[(long)r * ldg + c];
  }
#endif
}

__device__ __forceinline__ v8f vzero8() {
  v8f z;
#pragma unroll
  for (int i = 0; i < 8; ++i) z[i] = 0.0f;
  return z;
}

__device__ __forceinline__ v8f wmma_f32(v2f a, v2f b, v8f c) {
  // D = A(16x4) * B(4x16) + C, fp32 native
  return __builtin_amdgcn_wmma_f32_16x16x4_f32(false, a, false, b, (short)0, c,
                                               false, false);
}

// A-frag (16x4): lane<16 -> (row, k..k+1), lane>=16 -> (row, k+2..k+3); row-major src
__device__ __forceinline__ v2f afrag(const float* p, int row0, int k, int ld) {
  int lane = threadIdx.x & 31;
  const float* q = p + (long)(row0 + (lane & 15)) * ld + k + ((lane >> 4) << 1);
  v2f a;
  a.x = q[0];
  a.y = q[1];
  return a;
}

// B-frag (4x16): lane<16 -> rows k,k+1 at col; lane>=16 -> rows k+2,k+3; row-major src
__device__ __forceinline__ v2f bfrag(const float* p, int k, int col0, int ld) {
  int lane = threadIdx.x & 31;
  const float* q = p + (long)(k + ((lane >> 4) << 1)) * ld + col0 + (lane & 15);
  v2f b;
  b.x = q[0];
  b.y = q[ld];
  return b;
}

// C/D tile (16x16 f32): lane<16 holds rows 0..7 (VGPR v) at col=lane; lane>=16 rows 8..15
__device__ __forceinline__ void store_ctile(float* p, v8f c, int row0, int col0,
                                            long ld) {
  int lane = threadIdx.x & 31;
  long r = row0 + ((lane >> 4) << 3);
  float* q = p + r * ld + (col0 + (lane & 15));
#pragma unroll
  for (int v = 0; v < 8; ++v) q[v * ld] = c[v];
}

// ---------------------------------------------------------------------------
// Trig tables (DFT matrices), rebuilt every call (deterministic, graph-safe)
// ---------------------------------------------------------------------------
__global__ void __launch_bounds__(256) build_tables_k(float* tab) {
  int i = blockIdx.x * 256 + threadIdx.x;  // 0..16383
  if (i >= 16384) return;
  const float w = 6.283185307179586f / 128.0f;
  int r = i >> 7, c = i & 127;
  float ph = w * (float)((r * c) & 127);
  float cs = cosf(ph), sn = sinf(ph);
  tab[T_WCF + i] = cs;                      // forward column DFT cos
  tab[T_WSF + i] = sn;                      // forward column DFT sin
  tab[T_WCI + i] = cs * (1.0f / 128.0f);    // inverse (1/N folded)
  tab[T_WSI + i] = -sn * (1.0f / 128.0f);   // inverse, sign baked -> uniform kernel
  if (c < FP) {                             // Fy: [y=r][k=c], real rfft matrix
    int idx = r * FP + c;
    float re = 0.f, im = 0.f;
    if (c < NYR) { re = cs; im = -sn; }
    tab[T_FYRE + idx] = re;
    tab[T_FYIM + idx] = im;
  }
  if (r < FP) {                             // G: [k=r][y=c], irfft-to-real matrix
    int idx = r * 128 + c;
    float gr = 0.f, gi = 0.f;
    if (r < NYR) {
      float ck = (r == 0 || r == 64) ? 1.0f : 2.0f;
      gr = ck * cs * (1.0f / 128.0f);
      gi = -ck * sn * (1.0f / 128.0f);
    }
    tab[T_GR + idx] = gr;
    tab[T_GI + idx] = gi;
  }
}

// ---------------------------------------------------------------------------
// LayerNorm over channel dim (ddof=0, matches jnp.var)
// ---------------------------------------------------------------------------
__global__ void __launch_bounds__(256) ln_k(const float* __restrict__ x,
                                            const float* __restrict__ gamma,
                                            const float* __restrict__ beta,
                                            float* __restrict__ z) {
  int p = blockIdx.x * 256 + threadIdx.x;  // (b, pixel)
  int b = p >> 14;
  int r = p & 16383;
  const float* xb = x + (long)b * ND * NPIX + r;
  float s = 0.f, ss = 0.f;
  for (int d = 0; d < ND; ++d) {
    float v = xb[(long)d * NPIX];
    s += v;
    ss += v * v;
  }
  float mu = s * (1.0f / 128.0f);
  float var = ss * (1.0f / 128.0f) - mu * mu;
  float rs = rsqrtf(var + 1e-5f);
  float* zb = z + (long)b * ND * NPIX + r;
  for (int d = 0; d < ND; ++d) {
    float v = (xb[(long)d * NPIX] - mu) * rs;
    zb[(long)d * NPIX] = v * gamma[d] + beta[d];
  }
}

// ---------------------------------------------------------------------------
// Stage 1: row rfft as GEMM.  out[m, 0..79] = z[m, y] @ Fy[y, k]  (re & im)
// ---------------------------------------------------------------------------
__global__ void __launch_bounds__(256) rowfft_k(const float* __restrict__ z,
                                                const float* __restrict__ tab,
                                                float* __restrict__ outRe,
                                                float* __restrict__ outIm) {
  __shared__ float lA[128 * LDA132];
  __shared__ float lBr[128 * FP];
  __shared__ float lBi[128 * FP];
  int tid = threadIdx.x;
  long m0 = (long)blockIdx.x * 128;
  stage_tile(lA, LDA132, z + m0 * 128, 128, 128, 128, tid);
  stage_tile(lBr, FP, tab + T_FYRE, FP, 128, FP, tid);
  stage_tile(lBi, FP, tab + T_FYIM, FP, 128, FP, tid);
  async_join();
  __syncthreads();
  int w = tid >> 5, row0 = w << 4;
  v8f aRe[5], aIm[5];
#pragma unroll
  for (int n = 0; n < 5; ++n) { aRe[n] = vzero8(); aIm[n] = vzero8(); }
  for (int k = 0; k < 128; k += 4) {
    v2f a = afrag(lA, row0, k, LDA132);
#pragma unroll
    for (int n = 0; n < 5; ++n) {
      v2f br = bfrag(lBr, k, n * 16, FP);
      v2f bi = bfrag(lBi, k, n * 16, FP);
      aRe[n] = wmma_f32(a, br, aRe[n]);
      aIm[n] = wmma_f32(a, bi, aIm[n]);
    }
  }
#pragma unroll
  for (int n = 0; n < 5; ++n) {
    store_ctile(outRe + m0 * FP, aRe[n], row0, n * 16, FP);
    store_ctile(outIm + m0 * FP, aIm[n], row0, n * 16, FP);
  }
}

// ---------------------------------------------------------------------------
// Stages 2 & 4: complex 128-pt DFT along X as GEMM (per (b, channel) matrix).
//   Re' = Wc@Re + Ws@Im ;  Im' = Wc@Im - Ws@Re   (inverse sign baked into Ws table)
// ---------------------------------------------------------------------------
__global__ void __launch_bounds__(256) coldft_k(const float* __restrict__ inRe,
                                                const float* __restrict__ inIm,
                                                const float* __restrict__ wc,
                                                const float* __restrict__ wsn,
                                                float* __restrict__ outRe,
                                                float* __restrict__ outIm) {
  __shared__ float lRe[128 * FP];
  __shared__ float lIm[128 * FP];
  int tid = threadIdx.x;
  long base = (long)blockIdx.x * (128 * FP);
  stage_tile(lRe, FP, inRe + base, FP, 128, FP, tid);
  stage_tile(lIm, FP, inIm + base, FP, 128, FP, tid);
  async_join();
  __syncthreads();
  int w = tid >> 5, row0 = w << 4;
  v8f accRe[5], accIm[5];
#pragma unroll
  for (int n = 0; n < 5; ++n) { accRe[n] = vzero8(); accIm[n] = vzero8(); }
  for (int k = 0; k < 128; k += 4) {
    v2f ac = afrag(wc, row0, k, 128);   // trig A-frags straight from L2-hot table
    v2f as = afrag(wsn, row0, k, 128);
#pragma unroll
    for (int n = 0; n < 5; ++n) {
      v2f br = bfrag(lRe, k, n * 16, FP);
      v2f bi = bfrag(lIm, k, n * 16, FP);
      accRe[n] = wmma_f32(ac, br, accRe[n]);
      accRe[n] = wmma_f32(as, bi, accRe[n]);
      accIm[n] = wmma_f32(ac, bi, accIm[n]);
      v2f nbr;
      nbr.x = -br.x;
      nbr.y = -br.y;
      accIm[n] = wmma_f32(as, nbr, accIm[n]);
    }
  }
#pragma unroll
  for (int n = 0; n < 5; ++n) {
    store_ctile(outRe + base, accRe[n], row0, n * 16, FP);
    store_ctile(outIm + base, accIm[n], row0, n * 16, FP);
  }
}

// ---------------------------------------------------------------------------
// Stage 3: spectral channel mix.  For each filter l:
//   Y = Theta_l @ Xf   (two real GEMMs, X tile resident in LDS across all 16 l)
//   Sf += conj(Phi_l) * Y   (per-lane scalar combine: lane == N column)
// ---------------------------------------------------------------------------
__global__ void __launch_bounds__(256) spectral_k(const float* __restrict__ xRe,
                                                  const float* __restrict__ xIm,
                                                  const float* __restrict__ theta,
                                                  const float* __restrict__ phiRe,
                                                  const float* __restrict__ phiIm,
                                                  float* __restrict__ sRe,
                                                  float* __restrict__ sIm) {
  __shared__ float lTh[128 * LDA132];
  __shared__ float lRe[128 * FP];
  __shared__ float lIm[128 * FP];
  int tid = threadIdx.x;
  int b = blockIdx.x / 160;
  int col0 = (blockIdx.x % 160) * 64;
  long xbase = (long)b * ND * SPW;
  stage_tile(lRe, FP, xRe + xbase + col0, SPW, 128, 64, tid);  // 128 ch x 64 cols
  stage_tile(lIm, FP, xIm + xbase + col0, SPW, 128, 64, tid);
  async_join();
  int w = tid >> 5, lane = tid & 31, row0 = w << 4;
  v8f sAccRe[4], sAccIm[4];
#pragma unroll
  for (int n = 0; n < 4; ++n) { sAccRe[n] = vzero8(); sAccIm[n] = vzero8(); }
  for (int l = 0; l < NL; ++l) {
    __syncthreads();  // protect lTh from previous iteration readers
    stage_tile(lTh, LDA132, theta + (long)l * ND * ND, 128, 128, 128, tid);
    if (l + 1 < NL)  // prefetch next Theta tile (256 thr * 256B = 64KB)
      __builtin_prefetch(theta + ((long)(l + 1) * ND) * ND + tid * 64, 0, 1);
    async_join();
    __syncthreads();
    v8f yRe[4], yIm[4];
#pragma unroll
    for (int n = 0; n < 4; ++n) { yRe[n] = vzero8(); yIm[n] = vzero8(); }
    for (int k = 0; k < 128; k += 4) {
      v2f a = afrag(lTh, row0, k, LDA132);
#pragma unroll
      for (int n = 0; n < 4; ++n) {
        v2f br = bfrag(lRe, k, n * 16, FP);
        v2f bi = bfrag(lIm, k, n * 16, FP);
        yRe[n] = wmma_f32(a, br, yRe[n]);
        yIm[n] = wmma_f32(a, bi, yIm[n]);
      }
    }
#pragma unroll
    for (int n = 0; n < 4; ++n) {
      int c = col0 + n * 16 + (lane & 15);
      int kx = c / FP, kk = c % FP;
      float pr = 0.f, pi = 0.f;
      if (kk < NYR) {
        long pidx = ((long)l * NGX + kx) * NYR + kk;
        pr = phiRe[pidx];
        pi = phiIm[pidx];
      }
#pragma unroll
      for (int v = 0; v < 8; ++v) {
        sAccRe[n][v] += pr * yRe[n][v] + pi * yIm[n][v];
        sAccIm[n][v] += pr * yIm[n][v] - pi * yRe[n][v];
      }
    }
  }
  long obase = (long)b * ND * SPW;
#pragma unroll
  for (int n = 0; n < 4; ++n) {
    store_ctile(sRe + obase, sAccRe[n], row0, col0 + n * 16, SPW);
    store_ctile(sIm + obase, sAccIm[n], row0, col0 + n * 16, SPW);
  }
}

// ---------------------------------------------------------------------------
// Stage 5: inverse row transform to real.  S[m, y] = Re[m,k]@Gr[k,y] + Im[m,k]@Gi[k,y]
// ---------------------------------------------------------------------------
__global__ void __launch_bounds__(256) rowinv_k(const float* __restrict__ inRe,
                                                const float* __restrict__ inIm,
                                                const float* __restrict__ tab,
                                                float* __restrict__ outS) {
  __shared__ float lRe[128 * LDA84];
  __shared__ float lIm[128 * LDA84];
  __shared__ float lGr[FP * LDG144];
  __shared__ float lGi[FP * LDG144];
  int tid = threadIdx.x;
  long m0 = (long)blockIdx.x * 128;
  stage_tile(lRe, LDA84, inRe + m0 * FP, FP, 128, FP, tid);
  stage_tile(lIm, LDA84, inIm + m0 * FP, FP, 128, FP, tid);
  stage_tile(lGr, LDG144, tab + T_GR, 128, FP, 128, tid);
  stage_tile(lGi, LDG144, tab + T_GI, 128, FP, 128, tid);
  async_join();
  __syncthreads();
  int w = tid >> 5, row0 = w << 4;
  v8f acc[8];
#pragma unroll
  for (int n = 0; n < 8; ++n) acc[n] = vzero8();
  for (int k = 0; k < FP; k += 4) {
    v2f ar = afrag(lRe, row0, k, LDA84);
    v2f ai = afrag(lIm, row0, k, LDA84);
#pragma unroll
    for (int n = 0; n < 8; ++n) {
      v2f br = bfrag(lGr, k, n * 16, LDG144);
      v2f bi = bfrag(lGi, k, n * 16, LDG144);
      acc[n] = wmma_f32(ar, br, acc[n]);
      acc[n] = wmma_f32(ai, bi, acc[n]);
    }
  }
#pragma unroll
  for (int n = 0; n < 8; ++n)
    store_ctile(outS + m0 * 128, acc[n], row0, n * 16, 128);
}

// ---------------------------------------------------------------------------
// Fused 1x1-conv MLP: out = x + W2 @ gelu(W1 @ S + b1) + b2  (H stays in LDS)
// ---------------------------------------------------------------------------
__global__ void __launch_bounds__(256) mlp_k(const float* __restrict__ S,
                                             const float* __restrict__ x,
                                             const float* __restrict__ W1,
                                             const float* __restrict__ b1,
                                             const float* __restrict__ W2,
                                             const float* __restrict__ b2,
                                             float* __restrict__ out) {
  __shared__ float lS[128 * FP];  // K x N tile for GEMM1
  __shared__ float lH[256 * FP];  // K x N tile for GEMM2
  int tid = threadIdx.x;
  int b = blockIdx.x >> 8;
  int p0 = (blockIdx.x & 255) * 64;
  long base = (long)b * ND * NPIX;
  stage_tile(lS, FP, S + base + p0, NPIX, 128, 64, tid);
  async_join();
  __syncthreads();
  int w = tid >> 5, lane = tid & 31;
  // GEMM1: H(256 x 64); wave w owns M-rows [16w,16w+16) and [128+16w, ...)
  v8f h0[4], h1[4];
#pragma unroll
  for (int n = 0; n < 4; ++n) { h0[n] = vzero8(); h1[n] = vzero8(); }
  for (int k = 0; k < 128; k += 4) {
    v2f a0 = afrag(W1, w * 16, k, 128);
    v2f a1 = afrag(W1, 128 + w * 16, k, 128);
#pragma unroll
    for (int n = 0; n < 4; ++n) {
      v2f bs = bfrag(lS, k, n * 16, FP);
      h0[n] = wmma_f32(a0, bs, h0[n]);
      h1[n] = wmma_f32(a1, bs, h1[n]);
    }
  }
#pragma unroll
  for (int n = 0; n < 4; ++n) {
    int col = n * 16 + (lane & 15);
    int r0 = w * 16 + ((lane >> 4) << 3);
#pragma unroll
    for (int v = 0; v < 8; ++v) {
      float hv = h0[n][v] + b1[r0 + v];
      hv = 0.5f * hv * (1.0f + erff(hv * 0.7071067811865475f));
      lH[(r0 + v) * FP + col] = hv;
      float hw = h1[n][v] + b1[128 + r0 + v];
      hw = 0.5f * hw * (1.0f + erff(hw * 0.7071067811865475f));
      lH[(128 + r0 + v) * FP + col] = hw;
    }
  }
  __syncthreads();
  // GEMM2: out(128 x 64), K = 256
  v8f o[4];
#pragma unroll
  for (int n = 0; n < 4; ++n) o[n] = vzero8();
  for (int k = 0; k < 256; k += 4) {
    v2f a = afrag(W2, w * 16, k, 256);
#pragma unroll
    for (int n = 0; n < 4; ++n) {
      v2f bh = bfrag(lH, k, n * 16, FP);
      o[n] = wmma_f32(a, bh, o[n]);
    }
  }
#pragma unroll
  for (int n = 0; n < 4; ++n) {
    int col = p0 + n * 16 + (lane & 15);
    int r0 = w * 16 + ((lane >> 4) << 3);
#pragma unroll
    for (int v = 0; v < 8; ++v) {
      int ch = r0 + v;
      long a = base + (long)ch * NPIX + col;
      out[a] = o[n][v] + b2[ch] + x[a];
    }
  }
}

// ---------------------------------------------------------------------------
extern "C" void kernel_launch(void* const* d_in, const int* in_sizes, int n_in,
                              void* d_out, int out_size, void* d_ws, size_t ws_size,
                              hipStream_t stream) {
  (void)in_sizes; (void)n_in; (void)out_size; (void)ws_size;
  const float* x     = (const float*)d_in[0];
  const float* phiRe = (const float*)d_in[1];
  const float* phiIm = (const float*)d_in[2];
  const float* theta = (const float*)d_in[3];
  const float* gamma = (const float*)d_in[4];
  const float* beta  = (const float*)d_in[5];
  const float* W1    = (const float*)d_in[6];
  const float* b1    = (const float*)d_in[7];
  const float* W2    = (const float*)d_in[8];
  const float* b2    = (const float*)d_in[9];

  float* ws = (float*)d_ws;
  const size_t zElems = (size_t)NB * ND * NPIX;  // 8.39M (also reused for S)
  const size_t sElems = (size_t)NB * ND * SPW;   // 5.24M per spectral plane
  float* z   = ws;
  float* aRe = z + zElems;
  float* aIm = aRe + sElems;
  float* bRe = aIm + sElems;
  float* bIm = bRe + sElems;
  float* tab = bIm + sElems;

  build_tables_k<<<64, 256, 0, stream>>>(tab);
  ln_k<<<(NB * NPIX) / 256, 256, 0, stream>>>(x, gamma, beta, z);
  rowfft_k<<<512, 256, 0, stream>>>(z, tab, aRe, aIm);                             // z -> A
  coldft_k<<<512, 256, 0, stream>>>(aRe, aIm, tab + T_WCF, tab + T_WSF, bRe, bIm); // A -> B (fwd)
  spectral_k<<<640, 256, 0, stream>>>(bRe, bIm, theta, phiRe, phiIm, aRe, aIm);    // B -> A (Sf)
  coldft_k<<<512, 256, 0, stream>>>(aRe, aIm, tab + T_WCI, tab + T_WSI, bRe, bIm); // A -> B (inv)
  rowinv_k<<<512, 256, 0, stream>>>(bRe, bIm, tab, z);                             // B -> S (reuses z)
  mlp_k<<<1024, 256, 0, stream>>>(z, x, W1, b1, W2, b2, (float*)d_out);
}